// SGCModel_25795573580201
// MI455X (gfx1250) — compile-verified
//
#include <hip/hip_runtime.h>
#include <math.h>

#define N_NODES 100000
#define FDIM    512
#define FDIM4   128   // FDIM / 4
#define NCLS    40
#define ROWBLKS (N_NODES / 16)   // 6250

typedef float v2f __attribute__((ext_vector_type(2)));
typedef float v8f __attribute__((ext_vector_type(8)));

// ---------------- degree / normalization ----------------

__global__ void k_init_deg(float* __restrict__ deg) {
    int i = blockIdx.x * blockDim.x + threadIdx.x;
    if (i < N_NODES) deg[i] = 1.0f;   // self-loop contribution
}

__global__ void k_deg_edges(const int* __restrict__ dst, float* __restrict__ deg, int E) {
    int i = blockIdx.x * blockDim.x + threadIdx.x;
    if (i < E) atomicAdd(&deg[dst[i]], 1.0f);
}

__global__ void k_deg_to_dinv(float* __restrict__ deg) {
    int i = blockIdx.x * blockDim.x + threadIdx.x;
    if (i < N_NODES) {
        float d = deg[i];
        deg[i] = (d > 0.0f) ? rsqrtf(d) : 0.0f;
    }
}

// ---------------- SpMM round: out = dinv[i]^2 * in[i]  (self loop) ------------

__global__ void k_selfloop_init(const float4* __restrict__ in, float4* __restrict__ out,
                                const float* __restrict__ dinv, int nitems) {
    int t = blockIdx.x * blockDim.x + threadIdx.x;
    if (t >= nitems) return;
    int node = t >> 7;       // / FDIM4
    float di = dinv[node];
    float w  = di * di;
    float4 v = in[t];
    v.x *= w; v.y *= w; v.z *= w; v.w *= w;
    out[t] = v;
}

// ---------------- SpMM round: out[dst] += dinv[src]*dinv[dst] * in[src] -------
// one work-item = one float4 (4 features) of one edge

__global__ void k_edge_spmm(const int* __restrict__ src, const int* __restrict__ dst,
                            const float* __restrict__ dinv,
                            const float4* __restrict__ hin, float* __restrict__ hout,
                            int nitems) {
    int t = blockIdx.x * blockDim.x + threadIdx.x;
    if (t >= nitems) return;
    int e = t >> 7;          // edge index
    int f = t & 127;         // float4 index within row
    int s = src[e];
    int d = dst[e];
    float w = dinv[s] * dinv[d];
    float4 v = hin[s * FDIM4 + f];
    float* o = hout + (size_t)d * FDIM + (size_t)f * 4;
    atomicAdd(o + 0, w * v.x);
    atomicAdd(o + 1, w * v.y);
    atomicAdd(o + 2, w * v.z);
    atomicAdd(o + 3, w * v.w);
}

// ---------------- classifier GEMM (WMMA f32 16x16x4) + fused log_softmax ------
// 8 waves per block; each wave computes a 16-row block over all 40 classes
// (3 N-tiles; 3rd tile's columns >= 40 are computed from a clamped W row and
// discarded — column n of D depends only on B lanes with lane&15==n, so no
// masking of the loads is needed and EXEC stays full with clean b64 loads).

__global__ __launch_bounds__(256) void k_gemm_logsoftmax(const float* __restrict__ h,
                                                         const float* __restrict__ W,
                                                         const float* __restrict__ bias,
                                                         float* __restrict__ out) {
    const int wave = threadIdx.x >> 5;
    const int rb   = blockIdx.x * 8 + wave;
    if (rb >= ROWBLKS) return;          // whole wave exits; no partial EXEC at WMMA

    const int lane = threadIdx.x & 31;
    const int row0 = rb * 16;
    const int m    = lane & 15;
    const int half = lane >> 4;
    const int koff = half * 2;          // K sub-offset per 16x16x4 f32 layout
    const int n    = lane & 15;         // column within N-tile (B and C/D share this)

    const int  n2   = 32 + n;
    const bool v2ok = (n2 < NCLS);
    const int  n2c  = v2ok ? n2 : (NCLS - 1);   // clamp: safe row, cols >=40 discarded

    const float* Arow = h + (size_t)(row0 + m) * FDIM + koff;
    const float* B0   = W + (size_t)n        * FDIM + koff;
    const float* B1   = W + (size_t)(16 + n) * FDIM + koff;
    const float* B2   = W + (size_t)n2c      * FDIM + koff;

    v8f c0 = {}; v8f c1 = {}; v8f c2 = {};

    #pragma unroll 4
    for (int k = 0; k < FDIM; k += 4) {
        v2f a;  a.x  = Arow[k]; a.y  = Arow[k + 1];
        v2f b0; b0.x = B0[k];   b0.y = B0[k + 1];
        v2f b1; b1.x = B1[k];   b1.y = B1[k + 1];
        v2f b2; b2.x = B2[k];   b2.y = B2[k + 1];   // unconditional load, full EXEC
        c0 = __builtin_amdgcn_wmma_f32_16x16x4_f32(false, a, false, b0, (short)0, c0, false, false);
        c1 = __builtin_amdgcn_wmma_f32_16x16x4_f32(false, a, false, b1, (short)0, c1, false, false);
        c2 = __builtin_amdgcn_wmma_f32_16x16x4_f32(false, a, false, b2, (short)0, c2, false, false);
    }

    const float bias0 = bias[n];
    const float bias1 = bias[16 + n];
    const float bias2 = bias[n2c];
    const float NEG   = -1e30f;

    #pragma unroll
    for (int r = 0; r < 8; ++r) {
        // lane holds (row = r + half*8, col = n) of each N-tile
        float x0 = c0[r] + bias0;
        float x1 = c1[r] + bias1;
        float x2 = v2ok ? (c2[r] + bias2) : NEG;   // mask garbage columns here only

        float mx = fmaxf(x0, fmaxf(x1, x2));
        for (int off = 8; off >= 1; off >>= 1)
            mx = fmaxf(mx, __shfl_xor(mx, off, 16));   // 16-lane segment = one row

        float s = __expf(x0 - mx) + __expf(x1 - mx) + (v2ok ? __expf(x2 - mx) : 0.0f);
        for (int off = 8; off >= 1; off >>= 1)
            s += __shfl_xor(s, off, 16);

        float lse = mx + __logf(s);

        int row = row0 + r + half * 8;
        float* o = out + (size_t)row * NCLS;
        o[n]      = x0 - lse;
        o[16 + n] = x1 - lse;
        if (v2ok) o[n2] = x2 - lse;
    }
}

// ---------------- launcher ----------------

extern "C" void kernel_launch(void* const* d_in, const int* in_sizes, int n_in,
                              void* d_out, int out_size, void* d_ws, size_t ws_size,
                              hipStream_t stream) {
    const float* x  = (const float*)d_in[0];
    const int*   ei = (const int*)d_in[1];     // [2, E] row-major int32
    const float* W  = (const float*)d_in[2];   // [40, 512]
    const float* b  = (const float*)d_in[3];   // [40]
    float* out = (float*)d_out;

    const int E = in_sizes[1] / 2;             // 3,200,000
    const int* src = ei;
    const int* dst = ei + E;

    // workspace layout
    char*  ws   = (char*)d_ws;
    float* dinv = (float*)ws;                                    // N floats (deg -> dinv)
    float* h1   = (float*)(ws + (size_t)512 * 1024);             // N x 512
    float* h2   = h1 + (size_t)N_NODES * FDIM;                   // N x 512

    const int threads = 256;

    // degree + normalization
    k_init_deg  <<<(N_NODES + threads - 1) / threads, threads, 0, stream>>>(dinv);
    k_deg_edges <<<(E + threads - 1) / threads, threads, 0, stream>>>(dst, dinv, E);
    k_deg_to_dinv<<<(N_NODES + threads - 1) / threads, threads, 0, stream>>>(dinv);

    const int nodeItems = N_NODES * FDIM4;     // 12.8M float4 items
    const int edgeItems = E * FDIM4;           // 409.6M items (< 2^31)

    // round 1: h1 = (A+I)_norm * x
    k_selfloop_init<<<(nodeItems + threads - 1) / threads, threads, 0, stream>>>(
        (const float4*)x, (float4*)h1, dinv, nodeItems);
    k_edge_spmm<<<(edgeItems + threads - 1) / threads, threads, 0, stream>>>(
        src, dst, dinv, (const float4*)x, h1, edgeItems);

    // round 2: h2 = (A+I)_norm * h1
    k_selfloop_init<<<(nodeItems + threads - 1) / threads, threads, 0, stream>>>(
        (const float4*)h1, (float4*)h2, dinv, nodeItems);
    k_edge_spmm<<<(edgeItems + threads - 1) / threads, threads, 0, stream>>>(
        src, dst, dinv, (const float4*)h1, h2, edgeItems);

    // classifier + log_softmax (WMMA): 6250 row-blocks, 8 waves/block
    k_gemm_logsoftmax<<<(ROWBLKS + 7) / 8, 256, 0, stream>>>(h2, W, b, out);
}